// TripletAttention_15169824489913
// MI455X (gfx1250) — compile-verified
//
#include <hip/hip_runtime.h>
#include <cstdint>

#define B_  2
#define D_  256
#define D2  (D_*D_)
#define D3  (D_*D_*D_)
#define EPSN 1e-5f

typedef __attribute__((ext_vector_type(2))) float v2f;
typedef __attribute__((ext_vector_type(8))) float v8f;

// ---------------------------------------------------------------------------
// Kernel 1: reduce over channel dim C -> rhw_max/rhw_mean [B][H][W]
// grid = B*H blocks, 256 threads. Thread: w-quad (t&63)*4, c-phase t>>6.
// float4 loads: each wave instruction covers 512B contiguous.
// ---------------------------------------------------------------------------
__global__ void k_reduce_c(const float* __restrict__ x,
                           float* __restrict__ rmax, float* __restrict__ rmean) {
    int b = blockIdx.x >> 8, h = blockIdx.x & 255, t = threadIdx.x;
    int wq = t & 63, csub = t >> 6;
    __shared__ float pm[4*D_];
    __shared__ float ps[4*D_];
    const float* p = x + (size_t)b*D3 + (size_t)h*D_ + wq*4;
    float4 mx; mx.x = mx.y = mx.z = mx.w = -INFINITY;
    float4 sm; sm.x = sm.y = sm.z = sm.w = 0.f;
    #pragma unroll 4
    for (int ci = 0; ci < 64; ++ci) {
        int c = ci*4 + csub;
        float4 v = *(const float4*)(p + (size_t)c*D2);
        mx.x = fmaxf(mx.x, v.x); mx.y = fmaxf(mx.y, v.y);
        mx.z = fmaxf(mx.z, v.z); mx.w = fmaxf(mx.w, v.w);
        sm.x += v.x; sm.y += v.y; sm.z += v.z; sm.w += v.w;
    }
    int w0 = wq*4;
    pm[csub*D_ + w0+0] = mx.x; pm[csub*D_ + w0+1] = mx.y;
    pm[csub*D_ + w0+2] = mx.z; pm[csub*D_ + w0+3] = mx.w;
    ps[csub*D_ + w0+0] = sm.x; ps[csub*D_ + w0+1] = sm.y;
    ps[csub*D_ + w0+2] = sm.z; ps[csub*D_ + w0+3] = sm.w;
    __syncthreads();
    float m = -INFINITY, s = 0.f;
    #pragma unroll
    for (int q = 0; q < 4; ++q) { m = fmaxf(m, pm[q*D_ + t]); s += ps[q*D_ + t]; }
    size_t o = ((size_t)b*D_ + h)*D_ + t;
    rmax[o]  = m;
    rmean[o] = s * (1.f/D_);
}

// ---------------------------------------------------------------------------
// Kernel 2: per (b,c) plane: reduce over H -> rhc[b][c][w], over W -> rwc[b][h][c]
// grid = B*C blocks, 256 threads. Thread: w-quad (t&63)*4, h-phase t>>6.
// Each h-row is covered by 2 waves; 4-elem pre-reduce then 5-step shfl.
// ---------------------------------------------------------------------------
__global__ void k_reduce_hw(const float* __restrict__ x,
                            float* __restrict__ hc_max, float* __restrict__ hc_mean,
                            float* __restrict__ wc_max, float* __restrict__ wc_mean) {
    int b = blockIdx.x >> 8, c = blockIdx.x & 255, t = threadIdx.x;
    int wq = t & 63, hsub = t >> 6;
    int lane = t & 31;
    int half = (wq >> 5) & 1;
    __shared__ float rp_max[D_*2];
    __shared__ float rp_sum[D_*2];
    __shared__ float cp_max[4*D_];
    __shared__ float cp_sum[4*D_];
    const float* p = x + (size_t)b*D3 + (size_t)c*D2 + wq*4;
    float4 cmx; cmx.x = cmx.y = cmx.z = cmx.w = -INFINITY;
    float4 csm; csm.x = csm.y = csm.z = csm.w = 0.f;
    for (int hg = 0; hg < 64; ++hg) {
        int h = hg*4 + hsub;
        float4 v = *(const float4*)(p + (size_t)h*D_);
        cmx.x = fmaxf(cmx.x, v.x); cmx.y = fmaxf(cmx.y, v.y);
        cmx.z = fmaxf(cmx.z, v.z); cmx.w = fmaxf(cmx.w, v.w);
        csm.x += v.x; csm.y += v.y; csm.z += v.z; csm.w += v.w;
        float rm = fmaxf(fmaxf(v.x, v.y), fmaxf(v.z, v.w));
        float rs = (v.x + v.y) + (v.z + v.w);
        #pragma unroll
        for (int off = 16; off >= 1; off >>= 1) {
            rm = fmaxf(rm, __shfl_xor(rm, off, 32));
            rs += __shfl_xor(rs, off, 32);
        }
        if (lane == 0) { rp_max[h*2 + half] = rm; rp_sum[h*2 + half] = rs; }
    }
    int w0 = wq*4;
    cp_max[hsub*D_ + w0+0] = cmx.x; cp_max[hsub*D_ + w0+1] = cmx.y;
    cp_max[hsub*D_ + w0+2] = cmx.z; cp_max[hsub*D_ + w0+3] = cmx.w;
    cp_sum[hsub*D_ + w0+0] = csm.x; cp_sum[hsub*D_ + w0+1] = csm.y;
    cp_sum[hsub*D_ + w0+2] = csm.z; cp_sum[hsub*D_ + w0+3] = csm.w;
    __syncthreads();
    // thread t -> h = t : rwc[b][h][c]
    float m2 = fmaxf(rp_max[t*2], rp_max[t*2+1]);
    float s2 = rp_sum[t*2] + rp_sum[t*2+1];
    size_t o2 = ((size_t)b*D_ + t)*D_ + c;
    wc_max[o2]  = m2;
    wc_mean[o2] = s2 * (1.f/D_);
    // thread t -> w = t : rhc[b][c][w]
    float m3 = -INFINITY, s3 = 0.f;
    #pragma unroll
    for (int q = 0; q < 4; ++q) { m3 = fmaxf(m3, cp_max[q*D_ + t]); s3 += cp_sum[q*D_ + t]; }
    size_t o3 = ((size_t)b*D_ + c)*D_ + t;
    hc_max[o3]  = m3;
    hc_mean[o3] = s3 * (1.f/D_);
}

// ---------------------------------------------------------------------------
// Kernel 3: 7x7 2->1 conv (pad 3) + BN + sigmoid via V_WMMA_F32_16X16X4_F32.
// A = weights broadcast across M rows, B = im2col patches (one pixel per N col).
// D rows are all identical; lane L holds pixel (L%16) in acc[0].
// grid = 3(br)*B*rows*2(halves) blocks, 256 threads (8 waves x 16 pixels)
// ---------------------------------------------------------------------------
__device__ __forceinline__ float patchv(const float* __restrict__ zmax,
                                        const float* __restrict__ zmean,
                                        int k, int y, int xx0) {
    int ci = k / 49; int r = k - ci*49;
    int ky = r / 7;  int kx = r - ky*7;
    int yy = y + ky - 3;
    int xc = xx0 + kx - 3;
    bool ok = ((unsigned)yy < 256u) && ((unsigned)xc < 256u);
    const float* z = ci ? zmean : zmax;
    return ok ? z[yy*256 + xc] : 0.f;
}

__global__ void k_conv_gate(const float* __restrict__ red, float* __restrict__ smap,
    const float* __restrict__ wA, const float* __restrict__ gA, const float* __restrict__ bA,
    const float* __restrict__ mA, const float* __restrict__ vA,
    const float* __restrict__ wB, const float* __restrict__ gB, const float* __restrict__ bB,
    const float* __restrict__ mB, const float* __restrict__ vB,
    const float* __restrict__ wC, const float* __restrict__ gC, const float* __restrict__ bC,
    const float* __restrict__ mC, const float* __restrict__ vC) {
    int idx  = blockIdx.x;
    int half = idx & 1;
    int row  = (idx >> 1) & 255;
    int b    = (idx >> 9) & 1;
    int br   = idx >> 10;

    const float* wp = (br == 0) ? wA : ((br == 1) ? wB : wC);
    float gg = (br == 0) ? gA[0] : ((br == 1) ? gB[0] : gC[0]);
    float bb = (br == 0) ? bA[0] : ((br == 1) ? bB[0] : bC[0]);
    float mm = (br == 0) ? mA[0] : ((br == 1) ? mB[0] : mC[0]);
    float vv = (br == 0) ? vA[0] : ((br == 1) ? vB[0] : vC[0]);

    const float* zmax  = red + (size_t)br*2*B_*D2 + (size_t)b*D2;
    const float* zmean = red + (size_t)br*2*B_*D2 + (size_t)B_*D2 + (size_t)b*D2;
    float* outp = smap + (size_t)br*B_*D2 + (size_t)b*D2;

    __shared__ float wgt[100];
    int t = threadIdx.x;
    if (t < 100) wgt[t] = (t < 98) ? wp[t] : 0.f;
    __syncthreads();

    int lane = t & 31, wid = t >> 5;
    int n    = lane & 15;
    int hi2  = (lane >> 4) & 1;     // second half-wave -> K offset +2
    int px0  = half*128 + wid*16;
    int xx0  = px0 + n;

    v8f acc = {0.f,0.f,0.f,0.f,0.f,0.f,0.f,0.f};
    for (int kc = 0; kc < 25; ++kc) {
        int k0i = kc*4 + hi2*2;     // K index for vgpr0 (K = vgpr + 2*half)
        int k1i = k0i + 1;          // K index for vgpr1
        float a0 = wgt[k0i];        // wgt[98..99] are zero
        float a1 = wgt[k1i];
        int k0c = (k0i < 98) ? k0i : 97;
        int k1c = (k1i < 98) ? k1i : 97;
        float p0 = patchv(zmax, zmean, k0c, row, xx0);
        float p1 = patchv(zmax, zmean, k1c, row, xx0);
        float b0 = (k0i < 98) ? p0 : 0.f;
        float b1 = (k1i < 98) ? p1 : 0.f;
        v2f av = {a0, a1};
        v2f bv = {b0, b1};
        acc = __builtin_amdgcn_wmma_f32_16x16x4_f32(false, av, false, bv,
                                                    (short)0, acc, false, false);
    }
    float conv  = acc[0];                       // pixel (lane%16), all M rows equal
    float scale = gg * rsqrtf(vv + EPSN);
    float yv    = (conv - mm) * scale + bb;
    float s     = 1.f / (1.f + expf(-yv));
    if (lane < 16) outp[row*256 + px0 + n] = s;
}

// ---------------------------------------------------------------------------
// Kernel 4: out[b,i,j,k] = (x[b,i,j,k]*(sig_hw[b,j,k]+sig_hc[b,i,k])
//                           + x[b,k,i,j]*sig_wc[b,i,k]) / 3
// Transposed x tile staged into LDS via global_load_async_to_lds_b32 with a
// 257-float padded row stride. Compute loop is float4 over k (b128 vmem).
// grid = B * 256(i) * 8(k-chunks of 32), 256 threads
// ---------------------------------------------------------------------------
__global__ void k_combine(const float* __restrict__ x,
                          const float* __restrict__ smap,
                          float* __restrict__ out) {
    __shared__ float tile[32*257];              // tile[k_local][j], stride 257
    int idx = blockIdx.x;
    int kq = idx & 7;
    int i  = (idx >> 3) & 255;
    int b  = idx >> 11;
    int k0 = kq * 32;
    int t  = threadIdx.x;

    const float* sm_hw = smap + (size_t)b*D2;
    const float* sm_hc = smap + (size_t)B_*D2   + (size_t)b*D2;
    const float* sm_wc = smap + (size_t)2*B_*D2 + (size_t)b*D2;

    // async gather: tile[r][j=t] = x[b, k0+r, i, t]
    const float* base = x + (size_t)b*D3 + (size_t)k0*D2 + (size_t)i*D_;
    #pragma unroll 4
    for (int r = 0; r < 32; ++r) {
        unsigned lo = (unsigned)(size_t)(&tile[r*257 + t]);   // LDS byte address
        const float* ga = base + (size_t)r*D2 + t;
        asm volatile("global_load_async_to_lds_b32 %0, %1, off"
                     :: "v"(lo), "v"(ga) : "memory");
    }
    asm volatile("s_wait_asynccnt 0x0" ::: "memory");
    __syncthreads();

    int kq4 = t & 7;            // k-quad: lanes span 8 quads = 32 consecutive k
    int j0  = t >> 3;           // 0..31
    int kg0 = k0 + kq4*4;
    float4 shc = *(const float4*)(sm_hc + (size_t)i*D_ + kg0);
    float4 swc = *(const float4*)(sm_wc + (size_t)i*D_ + kg0);
    const float* xb = x   + (size_t)b*D3 + (size_t)i*D2 + kg0;
    float*       ob = out + (size_t)b*D3 + (size_t)i*D2 + kg0;
    #pragma unroll 2
    for (int jj = 0; jj < 8; ++jj) {
        int j = jj*32 + j0;
        float4 xn  = *(const float4*)(xb + (size_t)j*D_);
        float4 shw = *(const float4*)(sm_hw + (size_t)j*D_ + kg0);
        float4 xt;
        xt.x = tile[(kq4*4+0)*257 + j];         // banks (4kq4+l+j0)%64: no conflict
        xt.y = tile[(kq4*4+1)*257 + j];
        xt.z = tile[(kq4*4+2)*257 + j];
        xt.w = tile[(kq4*4+3)*257 + j];
        float4 o;
        o.x = (xn.x*(shw.x + shc.x) + xt.x*swc.x) * (1.f/3.f);
        o.y = (xn.y*(shw.y + shc.y) + xt.y*swc.y) * (1.f/3.f);
        o.z = (xn.z*(shw.z + shc.z) + xt.z*swc.z) * (1.f/3.f);
        o.w = (xn.w*(shw.w + shc.w) + xt.w*swc.w) * (1.f/3.f);
        *(float4*)(ob + (size_t)j*D_) = o;
    }
}

// ---------------------------------------------------------------------------
extern "C" void kernel_launch(void* const* d_in, const int* in_sizes, int n_in,
                              void* d_out, int out_size, void* d_ws, size_t ws_size,
                              hipStream_t stream) {
    (void)in_sizes; (void)n_in; (void)out_size; (void)ws_size;
    const float* x    = (const float*)d_in[0];
    const float* w_hw = (const float*)d_in[1];
    const float* g_hw = (const float*)d_in[2];
    const float* b_hw = (const float*)d_in[3];
    const float* m_hw = (const float*)d_in[4];
    const float* v_hw = (const float*)d_in[5];
    const float* w_hc = (const float*)d_in[6];
    const float* g_hc = (const float*)d_in[7];
    const float* b_hc = (const float*)d_in[8];
    const float* m_hc = (const float*)d_in[9];
    const float* v_hc = (const float*)d_in[10];
    const float* w_wc = (const float*)d_in[11];
    const float* g_wc = (const float*)d_in[12];
    const float* b_wc = (const float*)d_in[13];
    const float* m_wc = (const float*)d_in[14];
    const float* v_wc = (const float*)d_in[15];

    // ws layout (floats): red[3][2][B][256*256] then smap[3][B][256*256]
    float* red  = (float*)d_ws;
    float* smap = red + (size_t)3*2*B_*D2;

    float* rhw_max  = red + (size_t)0*B_*D2;
    float* rhw_mean = red + (size_t)1*B_*D2;
    float* rhc_max  = red + (size_t)2*B_*D2;
    float* rhc_mean = red + (size_t)3*B_*D2;
    float* rwc_max  = red + (size_t)4*B_*D2;
    float* rwc_mean = red + (size_t)5*B_*D2;

    k_reduce_c<<<B_*D_, 256, 0, stream>>>(x, rhw_max, rhw_mean);
    k_reduce_hw<<<B_*D_, 256, 0, stream>>>(x, rhc_max, rhc_mean, rwc_max, rwc_mean);
    k_conv_gate<<<3*B_*D_*2, 256, 0, stream>>>(red, smap,
        w_hw, g_hw, b_hw, m_hw, v_hw,
        w_hc, g_hc, b_hc, m_hc, v_hc,
        w_wc, g_wc, b_wc, m_wc, v_wc);
    k_combine<<<B_*D_*8, 256, 0, stream>>>(x, smap, (float*)d_out);
}